// DNA_protein_block_43336220016938
// MI455X (gfx1250) — compile-verified
//
#include <hip/hip_runtime.h>

typedef __attribute__((ext_vector_type(2))) float v2f;
typedef __attribute__((ext_vector_type(8))) float v8f;

#define STEP 12
#define MAXS 10.0f

// Per wave: 32 outputs (two 16-wide WMMA tiles).
// A (16x4) = [PWM_j; PWMrc_j; 0...], B (4x16) = 16 sliding DNA channel-vectors.
// D[0][n] = score_f(i0+n), D[1][n] = score_r(i0+n) accumulated over 12 taps.
__global__ __launch_bounds__(256) void pwm_scan_wmma_kernel(
    const float* __restrict__ dna,     // (B, L, 4)
    const float* __restrict__ concen,  // (B, L+1)
    const float* __restrict__ pwm,     // (12, 4)
    const float* __restrict__ pwmrc,   // (12, 4)
    const float* __restrict__ lamp,    // scalar
    float* __restrict__ out,           // (B, L+1)
    int L)
{
    const int lane = threadIdx.x & 31;
    const int wave = threadIdx.x >> 5;
    const int n    = lane & 15;        // column / row-in-half index
    const int half = lane >> 4;        // which K-pair this lane holds
    const int b    = blockIdx.y;
    const int i0   = blockIdx.x * 256 + wave * 32;   // first output of this wave
    const int Lo   = L - STEP + 1;                    // valid conv outputs
    const int Ltot = L + 1;                           // padded output length

    // ---- A matrix per tap: row 0 = PWM, row 1 = PWMrc, rows 2..15 = 0 ----
    // Lane layout (32-bit A 16x4): lanes 0-15 -> {K0,K1}, lanes 16-31 -> {K2,K3}.
    v2f aw[STEP];
#pragma unroll
    for (int j = 0; j < STEP; ++j) {
        v2f w = {0.0f, 0.0f};
        if (n == 0)      w = *(const v2f*)(pwm   + j * 4 + 2 * half);
        else if (n == 1) w = *(const v2f*)(pwmrc + j * 4 + 2 * half);
        aw[j] = w;
    }

    const float* dnab = dna + (size_t)b * L * 4;

    // Clamp tile base positions once; taps j=0..11 then stay in-bounds
    // (p + 11 <= L-1 whenever p <= Lo-1). Tail tiles read valid-but-unused
    // data that the epilogue masks to zero.
    int p0 = i0 + n;        if (p0 > Lo - 1) p0 = Lo - 1;
    int p1 = i0 + 16 + n;   if (p1 > Lo - 1) p1 = Lo - 1;
    const float* ptr0 = dnab + (size_t)p0 * 4 + 2 * half;
    const float* ptr1 = dnab + (size_t)p1 * 4 + 2 * half;

    // Preload all B operands (immediate offsets -> one load clause).
    v2f b0[STEP], b1[STEP];
#pragma unroll
    for (int j = 0; j < STEP; ++j) {
        b0[j] = *(const v2f*)(ptr0 + j * 4);
        b1[j] = *(const v2f*)(ptr1 + j * 4);
    }

    v8f c0 = {};   // tile 0: outputs i0 .. i0+15
    v8f c1 = {};   // tile 1: outputs i0+16 .. i0+31
#pragma unroll
    for (int j = 0; j < STEP; ++j) {
        c0 = __builtin_amdgcn_wmma_f32_16x16x4_f32(false, aw[j], false, b0[j],
                                                   (short)0, c0, false, false);
        c1 = __builtin_amdgcn_wmma_f32_16x16x4_f32(false, aw[j], false, b1[j],
                                                   (short)0, c1, false, false);
    }

    // c0/c1 valid data lives in lanes 0-15 (c[0]=score_f, c[1]=score_r).
    // Move tile 1 results into lanes 16-31 so each lane owns output i0+lane.
    float t1f = __shfl(c1[0], n, 32);
    float t1r = __shfl(c1[1], n, 32);
    float sf = (half == 0) ? c0[0] : t1f;
    float sr = (half == 0) ? c0[1] : t1r;

    const int i = i0 + lane;
    if (i < Ltot) {
        float r = 0.0f;
        if (i < Lo) {
            float ind = (sf > 0.0f || sr > 0.0f) ? 1.0f : 0.0f;
            float s   = fmaxf(sf, sr);
            float k   = __expf((s - MAXS) * lamp[0]) * ind;
            r = k * concen[(size_t)b * Ltot + i];
        }
        out[(size_t)b * Ltot + i] = r;   // pad region [Lo, L] -> 0
    }
}

extern "C" void kernel_launch(void* const* d_in, const int* in_sizes, int n_in,
                              void* d_out, int out_size, void* d_ws, size_t ws_size,
                              hipStream_t stream) {
    const float* DNA    = (const float*)d_in[0];  // (B, L, 4, 1)
    const float* concen = (const float*)d_in[1];  // (B, L+1, 1, 1)
    const float* PWM    = (const float*)d_in[2];  // (12, 4, 1, 1)
    const float* PWMrc  = (const float*)d_in[3];  // (12, 4, 1, 1)
    const float* lam    = (const float*)d_in[4];  // scalar

    // Derive B, L from flat sizes: S0 = B*L*4, S1 = B*(L+1) = B*L + B
    const int S0 = in_sizes[0];
    const int S1 = in_sizes[1];
    const int BL = S0 / 4;
    const int B  = S1 - BL;
    const int L  = BL / B;
    const int Ltot = L + 1;

    dim3 grid((Ltot + 255) / 256, B);
    pwm_scan_wmma_kernel<<<grid, 256, 0, stream>>>(DNA, concen, PWM, PWMrc, lam,
                                                   (float*)d_out, L);
}